// image_prj_25116968747133
// MI455X (gfx1250) — compile-verified
//
#include <hip/hip_runtime.h>
#include <math.h>

// CDNA5 (gfx1250) wave32 WMMA types
typedef __attribute__((ext_vector_type(2))) float v2f;
typedef __attribute__((ext_vector_type(8))) float v8f;

#define W 512
#define H 512
#define NANG 45

// floor(x + 0.5) in a single VOP1 (gfx12-family mnemonic). This is exactly
// the reference's nearest-neighbor rounding.
__device__ __forceinline__ int cvt_nearest_i32(float x) {
    int r;
    asm("v_cvt_nearest_i32_f32 %0, %1" : "=v"(r) : "v"(x));
    return r;
}

// One wave computes 16 consecutive X outputs for a fixed (b, angle).
// Lane layout follows the 32-bit 16x4 A-matrix layout:
//   lane = g*16 + m   (g = lane>>4, m = lane&15)
//   A VGPR0 -> K = 2g,  A VGPR1 -> K = 2g+1,  M = m
// Each K slot accumulates a 32-long Y chunk; 4 chained
// V_WMMA_F32_16X16X4_F32 with B = ones reduce all 512 Y terms in fp32.
__global__ void __launch_bounds__(256)
radon_prj_kernel(const float* __restrict__ img, float* __restrict__ out) {
    const int tid  = threadIdx.x;
    const int lane = tid & 31;
    const int wave = tid >> 5;
    const int g    = lane >> 4;   // lane group (selects K pair)
    const int m    = lane & 15;   // M row within tile

    const int a  = blockIdx.y;    // angle index
    const int b  = blockIdx.z;    // batch index
    const int x0 = blockIdx.x * 128 + wave * 16;  // X tile base for this wave
    const int X  = x0 + m;

    // Angle + tf.contrib.image rotation transform (matches reference)
    const float ang = (float)a * 4.0f * 3.14159265358979323846f / 180.0f;
    const float c = cosf(ang);
    const float s = sinf(ang);
    const float w1 = (float)(W - 1);
    const float h1 = (float)(H - 1);
    const float xo = (w1 - (c * w1 - s * h1)) * 0.5f;
    const float yo = (h1 - (s * w1 + c * h1)) * 0.5f;

    const float Xf = (float)X;
    const float px = fmaf(c, Xf, xo);   // x_in = px - s*Y
    const float py = fmaf(s, Xf, yo);   // y_in = py + c*Y
    const float ns = -s;

    const float* __restrict__ imb = img + (size_t)b * (W * H);

    v8f d = {};                                  // C/D accumulator (16x16 f32)
    v2f bones; bones.x = 1.0f; bones.y = 1.0f;   // B = ones (4x16)

    // 4 super-chunks of 128 Y each; within a super-chunk this lane's two
    // A slots (K=2g, K=2g+1) cover Y ranges [k*32, k*32+32).
    for (int sc = 0; sc < 4; ++sc) {
        const int yb0 = sc * 128 + (g * 2) * 32;   // K = 2g chunk start
        float a0 = 0.0f;
        float a1 = 0.0f;
        float yf = (float)yb0;                     // incremental float Y (exact)
#pragma unroll 8
        for (int j = 0; j < 32; ++j) {
            // slot 0: Y = yb0 + j
            {
                const int xi = cvt_nearest_i32(fmaf(ns, yf, px));
                const int yi = cvt_nearest_i32(fmaf(c,  yf, py));
                const bool valid = ((unsigned)xi < (unsigned)W) &
                                   ((unsigned)yi < (unsigned)H);
                const int lin = valid ? (yi * W + xi) : 0;
                const float v = imb[lin];
                a0 += valid ? v : 0.0f;
            }
            // slot 1: Y = yb0 + 32 + j
            {
                const float yf1 = yf + 32.0f;      // exact
                const int xi = cvt_nearest_i32(fmaf(ns, yf1, px));
                const int yi = cvt_nearest_i32(fmaf(c,  yf1, py));
                const bool valid = ((unsigned)xi < (unsigned)W) &
                                   ((unsigned)yi < (unsigned)H);
                const int lin = valid ? (yi * W + xi) : 0;
                const float v = imb[lin];
                a1 += valid ? v : 0.0f;
            }
            yf += 1.0f;
        }
        v2f amat; amat.x = a0; amat.y = a1;
        // D[m][n] += sum_k A[m][k]  (same value broadcast across all 16 N)
        d = __builtin_amdgcn_wmma_f32_16x16x4_f32(
                /*neg_a=*/false, amat,
                /*neg_b=*/false, bones,
                /*c_mod=*/(short)0, d,
                /*reuse_a=*/false, /*reuse_b=*/false);
    }

    // D layout: VGPR j holds M=j (lanes 0-15) and M=8+j (lanes 16-31).
    // Lane 0 stores rows 0..7 of this tile; lane 16 stores rows 8..15.
    if (m == 0) {
        float* o = out + ((size_t)b * NANG + (size_t)a) * W + x0 + g * 8;
        float4 lo; lo.x = d[0]; lo.y = d[1]; lo.z = d[2]; lo.w = d[3];
        float4 hi; hi.x = d[4]; hi.y = d[5]; hi.z = d[6]; hi.w = d[7];
        *(float4*)(o)     = lo;
        *(float4*)(o + 4) = hi;
    }
}

extern "C" void kernel_launch(void* const* d_in, const int* in_sizes, int n_in,
                              void* d_out, int out_size, void* d_ws, size_t ws_size,
                              hipStream_t stream) {
    (void)in_sizes; (void)n_in; (void)d_ws; (void)ws_size; (void)out_size;
    const float* x = (const float*)d_in[0];   // [8,512,512,1] fp32
    float* out = (float*)d_out;               // [8,45,512,1] fp32

    dim3 grid(4, NANG, 8);   // 4 X-tiles of 128, 45 angles, 8 batches
    dim3 block(256);         // 8 waves; each wave owns 16 X outputs
    radon_prj_kernel<<<grid, block, 0, stream>>>(x, out);
}